// JKNet_76579266888071
// MI455X (gfx1250) — compile-verified
//
#include <hip/hip_runtime.h>

typedef float v2f __attribute__((ext_vector_type(2)));
typedef float v8f __attribute__((ext_vector_type(8)));

#define HID 64

// Hardware no-return fp32 atomic add at device scope (coherent at global L2).
// Guarantees GLOBAL_ATOMIC_ADD_F32 instead of a possible CAS-loop fallback from
// the HIP header's arch whitelist.
__device__ __forceinline__ void gatomic_fadd(float* p, float v) {
  asm volatile("global_atomic_add_f32 %0, %1, off scope:SCOPE_DEV"
               :: "v"(p), "v"(v)
               : "memory");
}

__global__ void fill_kernel(float* __restrict__ p, float v, int n) {
  int i = blockIdx.x * blockDim.x + threadIdx.x;
  if (i < n) p[i] = v;
}

__global__ void deg_kernel(const long long* __restrict__ ei, float* __restrict__ deg, int E) {
  int e = blockIdx.x * blockDim.x + threadIdx.x;
  if (e < E) {
    int d = (int)ei[(long long)E + e];
    gatomic_fadd(&deg[d], 1.0f);
  }
}

__global__ void rsqrt_kernel(float* __restrict__ d, int n) {
  int i = blockIdx.x * blockDim.x + threadIdx.x;
  if (i < n) d[i] = rsqrtf(d[i]);
}

__global__ void norm_kernel(const long long* __restrict__ ei, const float* __restrict__ dinv,
                            float* __restrict__ norm, int E) {
  int e = blockIdx.x * blockDim.x + threadIdx.x;
  if (e < E) {
    int s = (int)ei[e];
    int d = (int)ei[(long long)E + e];
    norm[e] = dinv[s] * dinv[d];
  }
}

// Out[M,64] = A[M,K] @ W[K,64] in exact fp32 via V_WMMA_F32_16X16X4_F32.
// 4 waves per block; wave w owns rows [blk*64 + 16w, +16) x all 64 cols (4 N-tiles).
// A frag (16x4 f32): lanes 0-15 = M, v0 holds K+0 (lo half)/K+2 (hi half), v1 = K+1/K+3.
// B frag (4x16 f32): lane = N, v0 holds row K+0 (lo half)/K+2 (hi half), v1 = K+1/K+3.
__global__ __launch_bounds__(128) void gemm_wmma_f32(
    const float* __restrict__ A, const float* __restrict__ W, float* __restrict__ Out,
    int M, int K, int lda) {
  const int lane = threadIdx.x & 31;
  const int wave = threadIdx.x >> 5;
  const int half = lane >> 4;
  const int l16  = lane & 15;
  const int rowBase = blockIdx.x * 64 + wave * 16;

  int arow = rowBase + l16;
  if (arow >= M) arow = M - 1;          // clamp padding rows; their output is never stored
  const float* arowp = A + (long long)arow * lda;

  v8f c[4] = {};
  for (int k = 0; k < K; k += 4) {
    const int kk = k + 2 * half;
    v2f a = *(const v2f*)(arowp + kk);  // (K=kk, K=kk+1) for this lane-half
    const float* w0 = W + kk * HID + l16;
    const float* w1 = w0 + HID;
#pragma unroll
    for (int nt = 0; nt < 4; ++nt) {
      v2f b;
      b.x = w0[nt * 16];
      b.y = w1[nt * 16];
      c[nt] = __builtin_amdgcn_wmma_f32_16x16x4_f32(
          /*neg_a=*/false, a, /*neg_b=*/false, b,
          /*c_mod=*/(short)0, c[nt], /*reuse_a=*/false, /*reuse_b=*/false);
    }
  }
  // C/D layout: VGPR v -> rows (v, v+8) split across lane halves; lane%16 = N.
#pragma unroll
  for (int v = 0; v < 8; ++v) {
    int row = rowBase + v + 8 * half;
    if (row < M) {
      float* o = Out + (long long)row * HID + l16;
      o[0]  = c[0][v];
      o[16] = c[1][v];
      o[32] = c[2][v];
      o[48] = c[3][v];
    }
  }
}

// agg initialized with the self-loop contribution (single writer per element, no atomics).
__global__ void selfloop_kernel(const float* __restrict__ hW, const float* __restrict__ dinv,
                                float* __restrict__ agg, int n64) {
  int i = blockIdx.x * blockDim.x + threadIdx.x;
  if (i < n64) {
    float di = dinv[i >> 6];
    agg[i] = hW[i] * di * di;
  }
}

// One wave per edge; each lane moves 2 of the 64 features. L2-resident working set.
__global__ __launch_bounds__(256) void scatter_kernel(
    const long long* __restrict__ ei, const float* __restrict__ norm,
    const float* __restrict__ hW, float* __restrict__ agg, int E) {
  int e = blockIdx.x * 8 + (threadIdx.x >> 5);
  if (e >= E) return;
  int lane = threadIdx.x & 31;
  int s = (int)ei[e];
  int d = (int)ei[(long long)E + e];
  float w = norm[e];
  float2 v = *(const float2*)(hW + (long long)s * HID + lane * 2);
  float* dp = agg + (long long)d * HID + lane * 2;
  gatomic_fadd(dp,     v.x * w);
  gatomic_fadd(dp + 1, v.y * w);
}

// h = relu(agg + b); jk = max(jk, h). (jk zero-init is exact since every h >= 0.)
__global__ void bias_relu_jk_kernel(const float* __restrict__ agg, const float* __restrict__ b,
                                    float* __restrict__ h, float* __restrict__ jk, int n64) {
  int i = blockIdx.x * blockDim.x + threadIdx.x;
  if (i < n64) {
    float v = agg[i] + b[i & 63];
    v = fmaxf(v, 0.0f);
    h[i] = v;
    jk[i] = fmaxf(jk[i], v);
  }
}

// logits = jk @ fcW + fcb, then row-wise log_softmax. One thread per node.
__global__ __launch_bounds__(128) void fc_logsoftmax_kernel(
    const float* __restrict__ jk, const float* __restrict__ fcW,
    const float* __restrict__ fcb, float* __restrict__ out, int N) {
  int i = blockIdx.x * blockDim.x + threadIdx.x;
  if (i >= N) return;
  float acc[40];
#pragma unroll
  for (int c = 0; c < 40; ++c) acc[c] = fcb[c];
  const float* row = jk + (long long)i * HID;
  for (int k = 0; k < HID; ++k) {
    float xv = row[k];
    const float* wr = fcW + k * 40;
#pragma unroll
    for (int c = 0; c < 40; ++c) acc[c] = fmaf(xv, wr[c], acc[c]);
  }
  float m = acc[0];
#pragma unroll
  for (int c = 1; c < 40; ++c) m = fmaxf(m, acc[c]);
  float s = 0.0f;
#pragma unroll
  for (int c = 0; c < 40; ++c) s += expf(acc[c] - m);
  float lse = m + logf(s);
  float* o = out + (long long)i * 40;
#pragma unroll
  for (int c = 0; c < 40; ++c) o[c] = acc[c] - lse;
}

extern "C" void kernel_launch(void* const* d_in, const int* in_sizes, int n_in,
                              void* d_out, int out_size, void* d_ws, size_t ws_size,
                              hipStream_t stream) {
  const float* x        = (const float*)d_in[0];       // [N,128]
  const long long* ei   = (const long long*)d_in[1];   // [2,E] int64
  const float* W0       = (const float*)d_in[2];       // [128,64]
  const float* b0       = (const float*)d_in[3];       // [64]
  const float* W_rest   = (const float*)d_in[4];       // [5,64,64]
  const float* b_rest   = (const float*)d_in[5];       // [5,64]
  const float* fcW      = (const float*)d_in[6];       // [64,40]
  const float* fcb      = (const float*)d_in[7];       // [40]
  float* out            = (float*)d_out;               // [N,40]

  const int NFEAT  = 128;
  const int N      = in_sizes[0] / NFEAT;              // 50000
  const int E      = in_sizes[1] / 2;                  // 800000
  const int NLAYER = in_sizes[4] / (HID * HID) + 1;    // 6
  const int n64    = N * HID;

  char* ws = (char*)d_ws;
  size_t off = 0;
  auto wsalloc = [&](size_t bytes) -> float* {
    float* p = (float*)(ws + off);
    off += (bytes + 255) & ~(size_t)255;
    return p;
  };
  float* dinv = wsalloc((size_t)N * 4);        // deg, then deg^-1/2 in place
  float* norm = wsalloc((size_t)E * 4);        // per-edge normalization
  float* hbuf = wsalloc((size_t)n64 * 4);      // h / hW (reused in place)
  float* agg  = wsalloc((size_t)n64 * 4);      // scatter accumulator
  float* jk   = wsalloc((size_t)n64 * 4);      // JK running max
  (void)ws_size; (void)n_in; (void)out_size;

  // ---- GCN normalization (once, reused by all layers) ----
  fill_kernel<<<(N + 255) / 256, 256, 0, stream>>>(dinv, 1.0f, N);   // self-loop degree
  deg_kernel<<<(E + 255) / 256, 256, 0, stream>>>(ei, dinv, E);
  rsqrt_kernel<<<(N + 255) / 256, 256, 0, stream>>>(dinv, N);
  norm_kernel<<<(E + 255) / 256, 256, 0, stream>>>(ei, dinv, norm, E);
  fill_kernel<<<(n64 + 255) / 256, 256, 0, stream>>>(jk, 0.0f, n64);

  // ---- 6 GCN layers ----
  for (int l = 0; l < NLAYER; ++l) {
    const float* in = (l == 0) ? x  : hbuf;
    const int    K  = (l == 0) ? NFEAT : HID;
    const float* Wl = (l == 0) ? W0 : (W_rest + (size_t)(l - 1) * HID * HID);
    const float* bl = (l == 0) ? b0 : (b_rest + (size_t)(l - 1) * HID);

    gemm_wmma_f32<<<(N + 63) / 64, 128, 0, stream>>>(in, Wl, hbuf, N, K, K);
    selfloop_kernel<<<(n64 + 255) / 256, 256, 0, stream>>>(hbuf, dinv, agg, n64);
    scatter_kernel<<<(E + 7) / 8, 256, 0, stream>>>(ei, norm, hbuf, agg, E);
    bias_relu_jk_kernel<<<(n64 + 255) / 256, 256, 0, stream>>>(agg, bl, hbuf, jk, n64);
  }

  // ---- JK -> FC -> log_softmax ----
  fc_logsoftmax_kernel<<<(N + 127) / 128, 128, 0, stream>>>(jk, fcW, fcb, out, N);
}